// _SegmentGATLayer_43662637532065
// MI455X (gfx1250) — compile-verified
//
#include <hip/hip_runtime.h>
#include <hip/hip_bf16.h>

typedef __attribute__((ext_vector_type(2))) float v2f;
typedef __attribute__((ext_vector_type(8))) float v8f;

#define NEG_INF_F (-1e30f)

// Problem dims (fixed by the reference)
// B=8, N=256, T=128, D=256, H=8, hd=32, T*hd=4096, TOPK=32

// ---------------------------------------------------------------------------
// Generic fp32 WMMA GEMM:  C[M,N] = A[M,K] * B[K,N]   (B optionally W.T)
// Block tile 128x256, 8 waves as 2(M) x 4(N), each wave computes 64x64 via a
// 4x4 grid of V_WMMA_F32_16X16X4_F32 accumulators (16 WMMA per 8 b64 LDS
// fragment loads). K staged in LDS in kc=32 slabs:
//   As[128][36]      : pitch 36 -> 16B-aligned b128 stores, conflict-free
//                      b64 fragment reads (36r mod 64 distinct over 16 lanes)
//   BsP[16][256][2]  : K-pair interleaved so a B fragment {B[k],B[k+1]} is a
//                      single aligned ds_load_b64 (no packing moves)
//
// AMODE: 0 = row-major A (lda);  1 = gather from mixed [B,H,N,T,hd]
// BTRANS:0 = B[k][n]=Bp[k*ldb+n];1 = B[k][n]=Bp[n*ldb+k]  (i.e. X @ W.T)
// CMODE: 0 = plain row-major C (ldc)
//        1 = scatter to v layout [B,H,N,T,hd]
//        2 = out = resid + C, row-major 256 cols
// ---------------------------------------------------------------------------
template<int AMODE, int BTRANS, int CMODE>
__global__ __launch_bounds__(256)
void wmma_gemm_kernel(const float* __restrict__ Ap, const float* __restrict__ Bp,
                      float* __restrict__ Cp, const float* __restrict__ Rp,
                      int M, int N, int K, int lda, int ldb, int ldc,
                      long sA, long sB, long sC)
{
  __shared__ float As[128][36];
  __shared__ float BsP[16][256][2];

  const int tid  = threadIdx.x;
  const int lane = tid & 31;
  const int wave = tid >> 5;
  const int wm   = wave & 1;      // 2 M-chunks of 64 rows
  const int wn   = wave >> 1;     // 4 N-chunks of 64 cols
  const int m0   = blockIdx.x * 128;
  const int n0   = blockIdx.y * 256;
  const long z   = blockIdx.z;
  Ap += z * sA; Bp += z * sB; Cp += z * sC;

  const int hi = lane >> 4;       // lane half (K pair select / M half)
  const int lo = lane & 15;

  v8f acc[4][4];
  const v8f vzero = {0.f,0.f,0.f,0.f,0.f,0.f,0.f,0.f};
#pragma unroll
  for (int am = 0; am < 4; ++am)
#pragma unroll
    for (int bn = 0; bn < 4; ++bn) acc[am][bn] = vzero;

  for (int k0 = 0; k0 < K; k0 += 32) {
    // ---- stage A tile 128x32 : 1024 float4 loads, 4 per thread ----
#pragma unroll
    for (int it = 0; it < 4; ++it) {
      const int idx = tid + it * 256;
      const int r = idx >> 3, c4 = (idx & 7) << 2;
      const float* src;
      if (AMODE == 0) {
        src = &Ap[(long)(m0 + r) * lda + (k0 + c4)];
      } else {
        const int m = m0 + r;
        const int b = m >> 15, i = (m >> 7) & 255, t = m & 127;
        const int h = k0 >> 5;
        src = &Ap[(((long)(b * 8 + h) * 256 + i) * 128 + t) * 32 + c4];
      }
      const float4 v = *(const float4*)src;
      *(float4*)&As[r][c4] = v;               // (r*36+c4) mult of 4 -> 16B aligned
    }
    // ---- stage B tile 32x256 into K-pair interleaved layout ----
    if (BTRANS) {
      // contiguous along k: float4 spans k-pairs (2j, 2j+1)
#pragma unroll
      for (int it = 0; it < 8; ++it) {
        const int idx = tid + it * 256;
        const int c = idx >> 3, j4 = (idx & 7) << 2;
        const float4 v = *(const float4*)&Bp[(long)(n0 + c) * ldb + (k0 + j4)];
        *(v2f*)&BsP[(j4 >> 1) + 0][c][0] = v2f{v.x, v.y};
        *(v2f*)&BsP[(j4 >> 1) + 1][c][0] = v2f{v.z, v.w};
      }
    } else {
      // contiguous along n
#pragma unroll
      for (int it = 0; it < 8; ++it) {
        const int idx = tid + it * 256;
        const int r = idx >> 6, c4 = (idx & 63) << 2;
        const float4 v = *(const float4*)&Bp[(long)(k0 + r) * ldb + (n0 + c4)];
        const int kp = r >> 1, par = r & 1;
        BsP[kp][c4 + 0][par] = v.x;
        BsP[kp][c4 + 1][par] = v.y;
        BsP[kp][c4 + 2][par] = v.z;
        BsP[kp][c4 + 3][par] = v.w;
      }
    }
    // prefetch next K slab while this one computes
    if (k0 + 32 < K) {
      if (AMODE == 0)
        __builtin_prefetch(&Ap[(long)(m0 + (tid >> 1)) * lda + (k0 + 32)], 0, 0);
      if (BTRANS)
        __builtin_prefetch(&Bp[(long)(n0 + tid) * ldb + (k0 + 32)], 0, 0);
      else
        __builtin_prefetch(&Bp[(long)(k0 + 32 + (tid >> 3)) * ldb + n0 + (tid & 7) * 32], 0, 0);
    }
    __syncthreads();

    // ---- 8 x (K=4) WMMA steps: 8 b64 fragment loads -> 16 WMMA ----
#pragma unroll
    for (int kk = 0; kk < 32; kk += 4) {
      const int kb = kk + (hi << 1);
      const int kp = kb >> 1;
      v2f af[4], bf[4];
#pragma unroll
      for (int am = 0; am < 4; ++am) {
        const int row = wm * 64 + am * 16 + lo;
        af[am] = *(const v2f*)&As[row][kb];
      }
#pragma unroll
      for (int bn = 0; bn < 4; ++bn) {
        const int col = wn * 64 + bn * 16 + lo;
        bf[bn] = *(const v2f*)&BsP[kp][col][0];
      }
#pragma unroll
      for (int am = 0; am < 4; ++am)
#pragma unroll
        for (int bn = 0; bn < 4; ++bn)
          acc[am][bn] = __builtin_amdgcn_wmma_f32_16x16x4_f32(
              false, af[am], false, bf[bn], (short)0, acc[am][bn], false, false);
    }
    __syncthreads();
  }

  // ---- epilogue: C/D layout lanes0-15 -> M=r, lanes16-31 -> M=r+8 ----
#pragma unroll
  for (int am = 0; am < 4; ++am) {
#pragma unroll
    for (int bn = 0; bn < 4; ++bn) {
#pragma unroll
      for (int r = 0; r < 8; ++r) {
        const int row = m0 + wm * 64 + am * 16 + r + hi * 8;
        const int col = n0 + wn * 64 + bn * 16 + lo;
        const float val = acc[am][bn][r];
        if (CMODE == 0) {
          Cp[(long)row * ldc + col] = val;
        } else if (CMODE == 1) {
          // scatter to v[b,h,j,t,e]
          const int b = row >> 15, j = (row >> 7) & 255, t = row & 127;
          const int h = col >> 5,  e = col & 31;
          Cp[(((long)(b * 8 + h) * 256 + j) * 128 + t) * 32 + e] = val;
        } else {
          const long off = (long)row * 256 + col;
          Cp[off] = Rp[off] + val;   // residual add
        }
      }
    }
  }
}

// ---------------------------------------------------------------------------
// Attention: scores + bias, raw softmax mean, top-32 threshold, masked
// softmax, attn store + head-mean. One wave per row i; block = 8 waves;
// grid = (N/8, B). Loops h internally so head means need no atomics.
// ---------------------------------------------------------------------------
__global__ __launch_bounds__(256)
void attn_topk_kernel(const float* __restrict__ q, const float* __restrict__ kmat,
                      const float* __restrict__ badj, float* __restrict__ attnws,
                      float* __restrict__ attn_mean, float* __restrict__ raw_mean)
{
  __shared__ float ks[256][33];   // pitch 33: conflict-free row reads
  const int tid = threadIdx.x, lane = tid & 31, wave = tid >> 5;
  const int b = blockIdx.y;
  const int i = blockIdx.x * 8 + wave;
  const float scale = 0.17677669529663687f;   // 1/sqrt(32)

  float amean[8], rmean[8];
#pragma unroll
  for (int jj = 0; jj < 8; ++jj) { amean[jj] = 0.f; rmean[jj] = 0.f; }

  for (int h = 0; h < 8; ++h) {
    __syncthreads();
    // stage k[b, :, h*32:(h+1)*32] : one row per thread, b128 global loads
#pragma unroll
    for (int c4 = 0; c4 < 32; c4 += 4) {
      const float4 kv = *(const float4*)&kmat[((long)(b * 256 + tid)) * 256 + h * 32 + c4];
      ks[tid][c4 + 0] = kv.x; ks[tid][c4 + 1] = kv.y;
      ks[tid][c4 + 2] = kv.z; ks[tid][c4 + 3] = kv.w;
    }
    __syncthreads();

    float qr[32];
#pragma unroll
    for (int c4 = 0; c4 < 32; c4 += 4) {
      const float4 qv = *(const float4*)&q[((long)(b * 256 + i)) * 256 + h * 32 + c4];
      qr[c4 + 0] = qv.x; qr[c4 + 1] = qv.y; qr[c4 + 2] = qv.z; qr[c4 + 3] = qv.w;
    }

    // scores for this wave's row: 8 per lane, j = jj*32 + lane
    float s8[8];
#pragma unroll
    for (int jj = 0; jj < 8; ++jj) {
      const int j = jj * 32 + lane;
      float s = 0.f;
#pragma unroll
      for (int c = 0; c < 32; ++c) s += qr[c] * ks[j][c];
      s *= scale;
      const float ba = badj[((long)b * 256 + i) * 256 + j];
      s += __logf(fmaxf(ba, 1e-12f));          // BIAS_BASE = 1.0
      s8[jj] = s;
    }

    // ---- raw softmax (wave32 reduction) ----
    float mx = NEG_INF_F;
#pragma unroll
    for (int jj = 0; jj < 8; ++jj) mx = fmaxf(mx, s8[jj]);
#pragma unroll
    for (int o = 16; o > 0; o >>= 1) mx = fmaxf(mx, __shfl_xor(mx, o, 32));
    float sum = 0.f, e8[8];
#pragma unroll
    for (int jj = 0; jj < 8; ++jj) { e8[jj] = __expf(s8[jj] - mx); sum += e8[jj]; }
#pragma unroll
    for (int o = 16; o > 0; o >>= 1) sum += __shfl_xor(sum, o, 32);
    const float inv = 1.f / sum;
#pragma unroll
    for (int jj = 0; jj < 8; ++jj) rmean[jj] += e8[jj] * inv * 0.125f;

    // ---- top-32 threshold: 32 rounds of wave-max-and-remove ----
    float w8[8];
#pragma unroll
    for (int jj = 0; jj < 8; ++jj) w8[jj] = s8[jj];
    float kth = NEG_INF_F;
    for (int t = 0; t < 32; ++t) {
      float lm = NEG_INF_F;
#pragma unroll
      for (int jj = 0; jj < 8; ++jj) lm = fmaxf(lm, w8[jj]);
#pragma unroll
      for (int o = 16; o > 0; o >>= 1) lm = fmaxf(lm, __shfl_xor(lm, o, 32));
      kth = lm;
#pragma unroll
      for (int jj = 0; jj < 8; ++jj) if (w8[jj] >= lm) w8[jj] = NEG_INF_F;
    }

    // ---- masked softmax (keep >= kth, matches reference tie rule) ----
    float m8[8], mx2 = NEG_INF_F;
#pragma unroll
    for (int jj = 0; jj < 8; ++jj) {
      m8[jj] = (s8[jj] >= kth) ? s8[jj] : NEG_INF_F;
      mx2 = fmaxf(mx2, m8[jj]);
    }
#pragma unroll
    for (int o = 16; o > 0; o >>= 1) mx2 = fmaxf(mx2, __shfl_xor(mx2, o, 32));
    float sum2 = 0.f, a8[8];
#pragma unroll
    for (int jj = 0; jj < 8; ++jj) { a8[jj] = __expf(m8[jj] - mx2); sum2 += a8[jj]; }
#pragma unroll
    for (int o = 16; o > 0; o >>= 1) sum2 += __shfl_xor(sum2, o, 32);
    const float inv2 = 1.f / sum2;
#pragma unroll
    for (int jj = 0; jj < 8; ++jj) {
      const float a = a8[jj] * inv2;
      const int j = jj * 32 + lane;
      attnws[((long)(b * 8 + h) * 256 + i) * 256 + j] = a;
      amean[jj] += a * 0.125f;
    }
  }

#pragma unroll
  for (int jj = 0; jj < 8; ++jj) {
    const int j = jj * 32 + lane;
    const long off = ((long)b * 256 + i) * 256 + j;
    attn_mean[off] = amean[jj];
    raw_mean[off]  = rmean[jj];
  }
}

// ---------------------------------------------------------------------------
extern "C" void kernel_launch(void* const* d_in, const int* in_sizes, int n_in,
                              void* d_out, int out_size, void* d_ws, size_t ws_size,
                              hipStream_t stream) {
  const float* h_val    = (const float*)d_in[0];  // [8,256,128,256]
  const float* z_map    = (const float*)d_in[1];  // [8,256,256]
  const float* base_adj = (const float*)d_in[2];  // [8,256,256]
  const float* Wq       = (const float*)d_in[3];  // [256,256]
  const float* Wk       = (const float*)d_in[4];
  const float* Wv       = (const float*)d_in[5];
  const float* Wo       = (const float*)d_in[6];
  float* out = (float*)d_out;
  float* ws  = (float*)d_ws;

  // Workspace layout (floats)
  float* qw = ws;                   //   524,288  q  [B,N,D]
  float* kw = qw + 524288;          //   524,288  k  [B,N,D]
  float* aw = kw + 524288;          // 4,194,304  attn [B,H,N,N]
  float* vw = aw + 4194304;         // 67,108,864 v  [B,H,N,T,hd]
  float* mw = vw + 67108864;        // 67,108,864 mixed [B,H,N,T,hd]

  float* out_main = out;            // [B,N,T,D]
  float* out_attn = out + 67108864; // [B,N,N] attn.mean(heads)
  float* out_raw  = out_attn + 524288;

  const dim3 blk(256);

  // q = z_map @ Wq.T ; k = z_map @ Wk.T   (M=2048, N=256, K=256)
  wmma_gemm_kernel<0,1,0><<<dim3(16,1,1), blk, 0, stream>>>(
      z_map, Wq, qw, nullptr, 2048,256,256, 256,256,256, 0,0,0);
  wmma_gemm_kernel<0,1,0><<<dim3(16,1,1), blk, 0, stream>>>(
      z_map, Wk, kw, nullptr, 2048,256,256, 256,256,256, 0,0,0);

  // v = h_val @ Wv.T, scattered to [B,H,N,T,hd]   (M=262144)
  wmma_gemm_kernel<0,1,1><<<dim3(2048,1,1), blk, 0, stream>>>(
      h_val, Wv, vw, nullptr, 262144,256,256, 256,256,0, 0,0,0);

  // scores/softmax/top-k/attn + head means
  attn_topk_kernel<<<dim3(32,8,1), blk, 0, stream>>>(
      qw, kw, base_adj, aw, out_attn, out_raw);

  // mixed = attn @ v, batched over B*H=64   (M=256, N=4096, K=256)
  wmma_gemm_kernel<0,0,0><<<dim3(2,16,64), blk, 0, stream>>>(
      aw, vw, mw, nullptr, 256,4096,256, 256,4096,4096,
      65536L, 1048576L, 1048576L);

  // out = h_val + mixed(gathered) @ Wo.T   (M=262144)
  wmma_gemm_kernel<1,1,2><<<dim3(2048,1,1), blk, 0, stream>>>(
      mw, Wo, out_main, h_val, 262144,256,256, 0,256,256, 0,0,0);
}